// ImprovedGCN_36155034698037
// MI455X (gfx1250) — compile-verified
//
#include <hip/hip_runtime.h>

#define N_NODES 10000
#define NFEAT   1024
#define NHID    1024
#define NCLASS  256
#define NEDGE   160000
#define BN_EPS  1e-5f

typedef __attribute__((ext_vector_type(2))) float v2f;
typedef __attribute__((ext_vector_type(8))) float v8f;
typedef __attribute__((ext_vector_type(4))) int   i32x4;

#if defined(__AMDGCN__) && defined(__gfx1250__) && \
    __has_builtin(__builtin_amdgcn_global_load_async_to_lds_b32) && \
    __has_builtin(__builtin_amdgcn_global_load_async_to_lds_b128) && \
    __has_builtin(__builtin_amdgcn_s_wait_asynccnt)
#define USE_ASYNC_LDS 1
typedef __attribute__((address_space(1))) int   as1_i32;
typedef __attribute__((address_space(3))) int   as3_i32;
typedef __attribute__((address_space(1))) i32x4 as1_i32x4;
typedef __attribute__((address_space(3))) i32x4 as3_i32x4;
#else
#define USE_ASYNC_LDS 0
#endif

// out[M x NCOLS] = A[M x K] @ W[K x NCOLS] + bias
// 256 threads = 8 waves; workgroup tile 16 (M) x 128 (N); one 16x16 WMMA tile per wave.
template<int K, int NCOLS>
__global__ __launch_bounds__(256)
void gemm_bias_wmma(const float* __restrict__ A, const float* __restrict__ W,
                    const float* __restrict__ bias, float* __restrict__ out) {
  constexpr int KC = 32;
  __shared__ float As[16][KC + 1];   // +1 pad: lanes hit distinct banks on column reads
  __shared__ float Bs[KC][128];

  const int tid    = threadIdx.x;
  const int wave   = tid >> 5;
  const int lane   = tid & 31;
  const int laneLo = lane & 15;
  const int laneHi = lane >> 4;            // 0 or 1 (half-wave)
  const int m0     = blockIdx.x * 16;      // row tile base (10000 % 16 == 0)
  const int n0     = blockIdx.y * 128;     // col tile base
  const int wcol   = wave * 16;            // wave's column slice in LDS / tile

  v8f acc = {};
  for (int k0 = 0; k0 < K; k0 += KC) {
#if USE_ASYNC_LDS
    // CDNA5 async global->LDS path: no VGPR round-trip, tracked by ASYNCcnt.
    // A tile (16 x 32): one b32 per lane per issue (row pad breaks 16B alignment)
    {
      int idx = tid;
      #pragma unroll
      for (int i = 0; i < (16 * KC) / 256; ++i) {
        int r = idx / KC, c = idx % KC;
        __builtin_amdgcn_global_load_async_to_lds_b32(
            (as1_i32*)&A[(size_t)(m0 + r) * K + (k0 + c)],
            (as3_i32*)&As[r][c], 0, 0);
        idx += 256;
      }
    }
    // B tile (32 x 128): one b128 per lane per issue (rows are 16B aligned)
    {
      int idx = tid;
      #pragma unroll
      for (int i = 0; i < (KC * 128 / 4) / 256; ++i) {
        int r = idx >> 5, c4 = idx & 31;
        __builtin_amdgcn_global_load_async_to_lds_b128(
            (as1_i32x4*)&W[(size_t)(k0 + r) * NCOLS + (n0 + c4 * 4)],
            (as3_i32x4*)&Bs[r][c4 * 4], 0, 0);
        idx += 256;
      }
    }
    __builtin_amdgcn_s_wait_asynccnt(0);
    __syncthreads();
#else
    // Fallback: stage via VGPRs (coalesced)
    {
      int idx = tid;
      #pragma unroll
      for (int i = 0; i < (16 * KC) / 256; ++i) {
        int r = idx / KC, c = idx % KC;
        As[r][c] = A[(size_t)(m0 + r) * K + (k0 + c)];
        idx += 256;
      }
    }
    {
      int idx = tid;
      #pragma unroll
      for (int i = 0; i < (KC * 128) / 256; ++i) {
        int r = idx >> 7, c = idx & 127;
        Bs[r][c] = W[(size_t)(k0 + r) * NCOLS + (n0 + c)];
        idx += 256;
      }
    }
    __syncthreads();
#endif

    #pragma unroll
    for (int kk = 0; kk < KC; kk += 4) {
      // ISA f32 fragment layouts:
      // A 16x4: lanes 0-15 hold K=kk,kk+1; lanes 16-31 hold K=kk+2,kk+3 (M = laneLo)
      // B 4x16: lanes 0-15 hold rows kk,kk+1; lanes 16-31 rows kk+2,kk+3 (N = laneLo)
      const int kf = kk + 2 * laneHi;
      v2f a, b;
      a.x = As[laneLo][kf];
      a.y = As[laneLo][kf + 1];
      b.x = Bs[kf][wcol + laneLo];
      b.y = Bs[kf + 1][wcol + laneLo];
      acc = __builtin_amdgcn_wmma_f32_16x16x4_f32(
          /*neg_a=*/false, a, /*neg_b=*/false, b,
          /*c_mod=*/(short)0, acc, /*reuse_a=*/false, /*reuse_b=*/false);
    }
    __syncthreads();
  }

  // C/D layout: VGPR v -> row (v + 8*laneHi), col laneLo
  const int colg = n0 + wcol + laneLo;
  const float bv = bias[colg];
  #pragma unroll
  for (int v = 0; v < 8; ++v) {
    int row = m0 + v + 8 * laneHi;
    out[(size_t)row * NCOLS + colg] = acc[v] + bv;
  }
}

// agg[rows[e], f] += vals[e] * support[cols[e], f]   (atomic scatter-add)
template<int F>
__global__ __launch_bounds__(256)
void spmm_scatter(const float* __restrict__ support, const float* __restrict__ vals,
                  const int* __restrict__ rows, const int* __restrict__ cols,
                  float* __restrict__ agg) {
  long long idx = (long long)blockIdx.x * 256 + threadIdx.x;
  if (idx >= (long long)NEDGE * F) return;
  int e = (int)(idx / F);
  int f = (int)(idx % F);
  float v = vals[e];
  int r = rows[e], c = cols[e];
  atomicAdd(&agg[(size_t)r * F + f], v * support[(size_t)c * F + f]);
}

// per-column partial sum / sum-of-squares (coalesced: threads span columns)
template<int F>
__global__ __launch_bounds__(256)
void bn_stats(const float* __restrict__ h, float* __restrict__ sum, float* __restrict__ sq) {
  const int c  = blockIdx.x * 256 + threadIdx.x;     // column
  const int r0 = blockIdx.y * 250;                   // 10000 / 40 row chunks
  float s = 0.f, s2 = 0.f;
  for (int r = r0; r < r0 + 250; ++r) {
    float x = h[(size_t)r * F + c];
    s += x;
    s2 += x * x;
  }
  atomicAdd(&sum[c], s);
  atomicAdd(&sq[c], s2);
}

// in-place BN (biased variance) + optional ReLU
template<int F, bool RELU>
__global__ __launch_bounds__(256)
void bn_norm(float* __restrict__ h, const float* __restrict__ sum,
             const float* __restrict__ sq, const float* __restrict__ gamma,
             const float* __restrict__ beta) {
  long long idx = (long long)blockIdx.x * 256 + threadIdx.x;
  if (idx >= (long long)N_NODES * F) return;
  int c = (int)(idx % F);
  const float inv_n = 1.0f / (float)N_NODES;
  float mean = sum[c] * inv_n;
  float var  = sq[c] * inv_n - mean * mean;
  float sc   = gamma[c] * rsqrtf(var + BN_EPS);
  float y = (h[idx] - mean) * sc + beta[c];
  if (RELU) y = fmaxf(y, 0.f);
  h[idx] = y;
}

extern "C" void kernel_launch(void* const* d_in, const int* in_sizes, int n_in,
                              void* d_out, int out_size, void* d_ws, size_t ws_size,
                              hipStream_t stream) {
  (void)in_sizes; (void)n_in; (void)out_size; (void)ws_size;
  const float* x   = (const float*)d_in[0];
  const float* ev  = (const float*)d_in[1];
  const float* W1  = (const float*)d_in[2];
  const float* b1  = (const float*)d_in[3];
  const float* g1  = (const float*)d_in[4];
  const float* be1 = (const float*)d_in[5];
  const float* W2  = (const float*)d_in[6];
  const float* b2  = (const float*)d_in[7];
  const float* g2  = (const float*)d_in[8];
  const float* be2 = (const float*)d_in[9];
  const int* erows = (const int*)d_in[10];
  const int* ecols = (const int*)d_in[11];
  float* out = (float*)d_out;

  char* ws = (char*)d_ws;
  float* support = (float*)ws;                                      // N*NHID (reused for layer 2)
  float* agg1    = (float*)(ws + (size_t)N_NODES * NHID * 4);       // N*NHID (becomes h in place)
  float* stats   = (float*)(ws + (size_t)N_NODES * NHID * 8);
  float* sum1 = stats;
  float* sq1  = stats + NHID;
  float* sum2 = stats + 2 * NHID;
  float* sq2  = stats + 2 * NHID + NCLASS;

  (void)hipMemsetAsync(agg1, 0, (size_t)N_NODES * NHID * 4, stream);
  (void)hipMemsetAsync(out, 0, (size_t)N_NODES * NCLASS * 4, stream);
  (void)hipMemsetAsync(stats, 0, (2 * NHID + 2 * NCLASS) * 4, stream);

  // ---- layer 1 ----
  gemm_bias_wmma<NFEAT, NHID>
      <<<dim3(N_NODES / 16, NHID / 128), 256, 0, stream>>>(x, W1, b1, support);
  {
    long long total = (long long)NEDGE * NHID;
    spmm_scatter<NHID><<<(int)((total + 255) / 256), 256, 0, stream>>>(
        support, ev, erows, ecols, agg1);
  }
  bn_stats<NHID><<<dim3(NHID / 256, 40), 256, 0, stream>>>(agg1, sum1, sq1);
  {
    long long total = (long long)N_NODES * NHID;
    bn_norm<NHID, true><<<(int)((total + 255) / 256), 256, 0, stream>>>(
        agg1, sum1, sq1, g1, be1);
  }

  // ---- layer 2 ----
  gemm_bias_wmma<NHID, NCLASS>
      <<<dim3(N_NODES / 16, NCLASS / 128), 256, 0, stream>>>(agg1, W2, b2, support);
  {
    long long total = (long long)NEDGE * NCLASS;
    spmm_scatter<NCLASS><<<(int)((total + 255) / 256), 256, 0, stream>>>(
        support, ev, erows, ecols, out);
  }
  bn_stats<NCLASS><<<dim3(NCLASS / 256, 40), 256, 0, stream>>>(out, sum2, sq2);
  {
    long long total = (long long)N_NODES * NCLASS;
    bn_norm<NCLASS, false><<<(int)((total + 255) / 256), 256, 0, stream>>>(
        out, sum2, sq2, g2, be2);
  }
}